// GraphRespiratory_75788992905488
// MI455X (gfx1250) — compile-verified
//
#include <hip/hip_runtime.h>

// ---------------------------------------------------------------------------
// MI455X (gfx1250) GNN refinement.
// Memory-bound (~2 GB moved vs ~27 GFLOP): use full-precision f32 WMMA
// (V_WMMA_F32_16X16X4_F32) for the GEMM chains, 16-row tiles per wave32,
// LDS relayout between chained GEMMs, f32 global atomics for segment_sum.
// ---------------------------------------------------------------------------

typedef __attribute__((ext_vector_type(2))) float v2f;
typedef __attribute__((ext_vector_type(8))) float v8f;

static __device__ __forceinline__ v8f wmma4(v2f a, v2f b, v8f c) {
  // D(16x16,f32) = A(16x4,f32) * B(4x16,f32) + C  — full fp32 precision.
  return __builtin_amdgcn_wmma_f32_16x16x4_f32(false, a, false, b, (short)0, c,
                                               false, false);
}

// A-fragment layout (wave32, f32 16x4): lane L -> row M = L&15;
// lanes 0-15 hold K=(k0+0,k0+1) in (x,y); lanes 16-31 hold K=(k0+2,k0+3).
// B-fragment (4x16): lanes 0-15 -> K rows k0+0 (x), k0+1 (y), N = L&15;
// lanes 16-31 -> K rows k0+2 (x), k0+3 (y), N = L-16.
// C/D (16x16): vgpr r -> M = r + (L>=16 ? 8 : 0), N = L&15.

#define LDS_LD 33  // 16x33 padded tile: conflict-free row & column access

// Load A fragments for a K=32 operand from a padded LDS tile.
static __device__ __forceinline__ void load_afrag_lds(const float* T, int lane,
                                                      v2f afr[8]) {
  const int m = lane & 15;
  const int koff = (lane >> 4) << 1;
#pragma unroll
  for (int kc = 0; kc < 8; ++kc) {
    int kk = kc * 4 + koff;
    afr[kc].x = T[m * LDS_LD + kk];
    afr[kc].y = T[m * LDS_LD + kk + 1];
  }
}

// D = A(16x32) @ W(32x32 row-major) + bias  (two 16-col N tiles)
static __device__ __forceinline__ void gemm_k32(const v2f afr[8],
                                                const float* __restrict__ W,
                                                const float* __restrict__ B,
                                                int lane, v8f d[2]) {
  const int n = lane & 15;
  const int koff = (lane >> 4) << 1;
#pragma unroll
  for (int nt = 0; nt < 2; ++nt) {
    const float* Wn = W + nt * 16;
    float bias = B[nt * 16 + n];
    v8f c = {bias, bias, bias, bias, bias, bias, bias, bias};
#pragma unroll
    for (int kc = 0; kc < 8; ++kc) {
      int kr = kc * 4 + koff;
      v2f b;
      b.x = Wn[kr * 32 + n];
      b.y = Wn[(kr + 1) * 32 + n];
      c = wmma4(afr[kc], b, c);
    }
    d[nt] = c;
  }
}

// Scatter a D-layout accumulator pair into a row-major padded LDS tile.
static __device__ __forceinline__ void store_d_lds(float* T, int lane,
                                                   const v8f d[2], bool relu) {
  const int n = lane & 15;
  const int mb = (lane >> 4) * 8;
#pragma unroll
  for (int nt = 0; nt < 2; ++nt) {
#pragma unroll
    for (int r = 0; r < 8; ++r) {
      float v = d[nt][r];
      if (relu) v = fmaxf(v, 0.0f);
      T[(mb + r) * LDS_LD + nt * 16 + n] = v;
    }
  }
}

// ---------------------------------------------------------------------------
// node_feats = node_logits @ nip_w + nip_b   (K=7 — pure streaming, VALU)
__global__ void node_init_kernel(const float* __restrict__ nl,
                                 const float* __restrict__ w,
                                 const float* __restrict__ b,
                                 float* __restrict__ nf, int n_nodes) {
  int t = blockIdx.x * blockDim.x + threadIdx.x;
  if (t >= n_nodes * 8) return;
  int node = t >> 3, c0 = (t & 7) * 4;
  const float* x = nl + (size_t)node * 7;
  float acc[4];
#pragma unroll
  for (int j = 0; j < 4; ++j) acc[j] = b[c0 + j];
#pragma unroll
  for (int k = 0; k < 7; ++k) {
    float xv = x[k];
#pragma unroll
    for (int j = 0; j < 4; ++j) acc[j] += xv * w[k * 32 + c0 + j];
  }
  float4 o = {acc[0], acc[1], acc[2], acc[3]};
  *(float4*)(nf + (size_t)node * 32 + c0) = o;
}

// ---------------------------------------------------------------------------
// edge_feats = relu(edge_logits @ em_w1 + b1) @ em_w2 + b2    (WMMA stage 2)
__global__ __launch_bounds__(32) void edge_init_kernel(
    const float* __restrict__ el, float* __restrict__ ef,
    const float* __restrict__ w1, const float* __restrict__ b1,
    const float* __restrict__ w2, const float* __restrict__ b2, int n_edges) {
  __shared__ float T[16 * LDS_LD];
  const int lane = threadIdx.x;
  const int m = lane & 15;
  const int cb = (lane >> 4) * 16;
  const long tile = (long)blockIdx.x * 16;
  long er = tile + m;
  if (er >= n_edges) er = n_edges - 1;
  float e0 = el[er * 2], e1 = el[er * 2 + 1];
#pragma unroll
  for (int j = 0; j < 16; ++j) {
    int c = cb + j;
    T[m * LDS_LD + c] = fmaxf(e0 * w1[c] + e1 * w1[32 + c] + b1[c], 0.0f);
  }
  __syncthreads();  // single-wave WG: barrier lowers to NOP + LDS ordering
  v2f afr[8];
  v8f d[2];
  load_afrag_lds(T, lane, afr);
  gemm_k32(afr, w2, b2, lane, d);
  __syncthreads();
  store_d_lds(T, lane, d, false);
  __syncthreads();
  if (tile + m < n_edges) {
    size_t eb = (size_t)(tile + m) * 32;
#pragma unroll
    for (int j = 0; j < 16; ++j) ef[eb + cb + j] = T[m * LDS_LD + cb + j];
  }
}

// ---------------------------------------------------------------------------
// Per-iteration edge pass:
//   t = [nf[src] | nf[dst]] @ ap_w + ap_b ; u = t@wv+bv ; v = u@wo+bo
//   ef = relu(ef + v) ; msg = relu(nf[src] + ef) ; agg[dst] += msg (atomics)
__global__ __launch_bounds__(32) void edge_pass_kernel(
    const float* __restrict__ nf, float* __restrict__ ef,
    const int* __restrict__ src, const int* __restrict__ dst,
    float* __restrict__ agg, const float* __restrict__ ap_w,
    const float* __restrict__ ap_b, const float* __restrict__ wv,
    const float* __restrict__ bv, const float* __restrict__ wo,
    const float* __restrict__ bo, int n_edges) {
  __shared__ float T[16 * LDS_LD];
  const int lane = threadIdx.x;
  const int m = lane & 15;
  const int koff = (lane >> 4) << 1;
  const int n = lane & 15;
  const long tile = (long)blockIdx.x * 16;
  long er = tile + m;
  if (er >= n_edges) er = n_edges - 1;
  const int si = src[er];
  const int di = dst[er];
  const float* rowS = nf + (size_t)si * 32;
  const float* rowD = nf + (size_t)di * 32;

  // Gather A fragments for the concatenated [src|dst] operand (K=64).
  v2f aS[8], aD[8];
#pragma unroll
  for (int kc = 0; kc < 8; ++kc) {
    int kk = kc * 4 + koff;
    aS[kc].x = rowS[kk];
    aS[kc].y = rowS[kk + 1];
    aD[kc].x = rowD[kk];
    aD[kc].y = rowD[kk + 1];
  }

  // Stage 1: K=64 attn_proj (rows 0-31 of ap_w hit src half, 32-63 dst half)
  v8f d[2];
#pragma unroll
  for (int nt = 0; nt < 2; ++nt) {
    const float* Wn = ap_w + nt * 16;
    float bias = ap_b[nt * 16 + n];
    v8f c = {bias, bias, bias, bias, bias, bias, bias, bias};
#pragma unroll
    for (int kc = 0; kc < 8; ++kc) {
      int kr = kc * 4 + koff;
      v2f b;
      b.x = Wn[kr * 32 + n];
      b.y = Wn[(kr + 1) * 32 + n];
      c = wmma4(aS[kc], b, c);
    }
#pragma unroll
    for (int kc = 0; kc < 8; ++kc) {
      int kr = 32 + kc * 4 + koff;
      v2f b;
      b.x = Wn[kr * 32 + n];
      b.y = Wn[(kr + 1) * 32 + n];
      c = wmma4(aD[kc], b, c);
    }
    d[nt] = c;
  }
  store_d_lds(T, lane, d, false);
  __syncthreads();

  // Stage 2: v-proj
  v2f afr[8];
  load_afrag_lds(T, lane, afr);
  gemm_k32(afr, wv, bv, lane, d);
  __syncthreads();
  store_d_lds(T, lane, d, false);
  __syncthreads();

  // Stage 3: out-proj
  load_afrag_lds(T, lane, afr);
  gemm_k32(afr, wo, bo, lane, d);
  __syncthreads();
  store_d_lds(T, lane, d, false);
  __syncthreads();

  // Epilogue: lane handles row m, 16 contiguous columns.
  const int cb = (lane >> 4) * 16;
  if (tile + m < n_edges) {
    size_t eb = (size_t)(tile + m) * 32;
    float* ag = agg + (size_t)di * 32;
#pragma unroll
    for (int j = 0; j < 16; ++j) {
      int c = cb + j;
      float v = T[m * LDS_LD + c];                // GAMMA == 1.0
      float en = fmaxf(ef[eb + c] + v, 0.0f);
      ef[eb + c] = en;
      float msg = fmaxf(rowS[c] + en, 0.0f);
      atomicAdd(&ag[c], msg);                     // global_atomic_add_f32
    }
  }
}

// ---------------------------------------------------------------------------
// node_feats = relu(relu((nf+agg)@g_w1+g_b1)@g_w2+g_b2)   (in-place, WMMA)
__global__ __launch_bounds__(32) void node_mlp_kernel(
    float* __restrict__ nf, const float* __restrict__ agg,
    const float* __restrict__ w1, const float* __restrict__ b1,
    const float* __restrict__ w2, const float* __restrict__ b2, int n_nodes) {
  __shared__ float T[16 * LDS_LD];
  const int lane = threadIdx.x;
  const int m = lane & 15;
  const int koff = (lane >> 4) << 1;
  const long tile = (long)blockIdx.x * 16;
  long nr = tile + m;
  if (nr >= n_nodes) nr = n_nodes - 1;
  const float* rN = nf + (size_t)nr * 32;
  const float* rA = agg + (size_t)nr * 32;
  v2f afr[8];
#pragma unroll
  for (int kc = 0; kc < 8; ++kc) {
    int kk = kc * 4 + koff;
    afr[kc].x = rN[kk] + rA[kk];
    afr[kc].y = rN[kk + 1] + rA[kk + 1];
  }
  v8f d[2];
  gemm_k32(afr, w1, b1, lane, d);
  store_d_lds(T, lane, d, true);  // relu
  __syncthreads();
  load_afrag_lds(T, lane, afr);
  gemm_k32(afr, w2, b2, lane, d);
  __syncthreads();
  store_d_lds(T, lane, d, true);  // final relu
  __syncthreads();
  const int cb = (lane >> 4) * 16;
  if (tile + m < n_nodes) {
    size_t nb = (size_t)(tile + m) * 32;
#pragma unroll
    for (int j = 0; j < 16; ++j) nf[nb + cb + j] = T[m * LDS_LD + cb + j];
  }
}

// ---------------------------------------------------------------------------
__global__ void zero_kernel(float* __restrict__ p, long nelems) {
  long t = (long)blockIdx.x * blockDim.x + threadIdx.x;
  if (t < nelems) p[t] = 0.0f;
}

__global__ void node_head_kernel(const float* __restrict__ nf,
                                 const float* __restrict__ w,
                                 const float* __restrict__ b,
                                 float* __restrict__ out, int n_nodes) {
  int t = blockIdx.x * blockDim.x + threadIdx.x;
  if (t >= n_nodes) return;
  const float* r = nf + (size_t)t * 32;
  float acc[7];
#pragma unroll
  for (int c = 0; c < 7; ++c) acc[c] = b[c];
#pragma unroll
  for (int k = 0; k < 32; ++k) {
    float rv = r[k];
#pragma unroll
    for (int c = 0; c < 7; ++c) acc[c] += rv * w[k * 7 + c];
  }
#pragma unroll
  for (int c = 0; c < 7; ++c) out[(size_t)t * 7 + c] = acc[c];
}

__global__ void edge_head_kernel(const float* __restrict__ ef,
                                 const float* __restrict__ w,
                                 const float* __restrict__ b,
                                 float* __restrict__ out, int n_edges) {
  int t = blockIdx.x * blockDim.x + threadIdx.x;
  if (t >= n_edges) return;
  const float* r = ef + (size_t)t * 32;
  float a0 = b[0], a1 = b[1];
#pragma unroll
  for (int k = 0; k < 32; ++k) {
    float rv = r[k];
    a0 += rv * w[k * 2];
    a1 += rv * w[k * 2 + 1];
  }
  out[(size_t)t * 2] = a0;
  out[(size_t)t * 2 + 1] = a1;
}

// ---------------------------------------------------------------------------
extern "C" void kernel_launch(void* const* d_in, const int* in_sizes, int n_in,
                              void* d_out, int out_size, void* d_ws,
                              size_t ws_size, hipStream_t stream) {
  const float* node_logits = (const float*)d_in[0];
  const float* edge_logits = (const float*)d_in[1];
  const int* edge_index = (const int*)d_in[2];
  const float* nip_w = (const float*)d_in[3];
  const float* nip_b = (const float*)d_in[4];
  const float* em_w1 = (const float*)d_in[5];
  const float* em_b1 = (const float*)d_in[6];
  const float* em_w2 = (const float*)d_in[7];
  const float* em_b2 = (const float*)d_in[8];
  const float* ap_w = (const float*)d_in[9];
  const float* ap_b = (const float*)d_in[10];
  const float* wv = (const float*)d_in[11];
  const float* bv = (const float*)d_in[12];
  const float* wo = (const float*)d_in[13];
  const float* bo = (const float*)d_in[14];
  const float* g_w1 = (const float*)d_in[15];
  const float* g_b1 = (const float*)d_in[16];
  const float* g_w2 = (const float*)d_in[17];
  const float* g_b2 = (const float*)d_in[18];
  const float* nch_w = (const float*)d_in[19];
  const float* nch_b = (const float*)d_in[20];
  const float* ech_w = (const float*)d_in[21];
  const float* ech_b = (const float*)d_in[22];

  const int n_nodes = in_sizes[0] / 7;
  const int n_edges = in_sizes[1] / 2;
  const int* src = edge_index;
  const int* dst = edge_index + n_edges;

  // Workspace carve: nf (N*32) | agg (N*32) | ef (E*32)  ~230 MB total
  float* nf = (float*)d_ws;
  float* agg = nf + (size_t)n_nodes * 32;
  float* ef = agg + (size_t)n_nodes * 32;

  float* out_node = (float*)d_out;
  float* out_edge = out_node + (size_t)n_nodes * 7;

  const int etiles = (n_edges + 15) / 16;
  const int ntiles = (n_nodes + 15) / 16;

  node_init_kernel<<<(n_nodes * 8 + 255) / 256, 256, 0, stream>>>(
      node_logits, nip_w, nip_b, nf, n_nodes);
  edge_init_kernel<<<etiles, 32, 0, stream>>>(edge_logits, ef, em_w1, em_b1,
                                              em_w2, em_b2, n_edges);

  for (int it = 0; it < 2; ++it) {
    const long aggN = (long)n_nodes * 32;
    zero_kernel<<<(int)((aggN + 255) / 256), 256, 0, stream>>>(agg, aggN);
    edge_pass_kernel<<<etiles, 32, 0, stream>>>(nf, ef, src, dst, agg, ap_w,
                                                ap_b, wv, bv, wo, bo, n_edges);
    node_mlp_kernel<<<ntiles, 32, 0, stream>>>(nf, agg, g_w1, g_b1, g_w2, g_b2,
                                               n_nodes);
  }

  node_head_kernel<<<(n_nodes + 255) / 256, 256, 0, stream>>>(nf, nch_w, nch_b,
                                                              out_node, n_nodes);
  edge_head_kernel<<<(n_edges + 255) / 256, 256, 0, stream>>>(ef, ech_w, ech_b,
                                                              out_edge, n_edges);
}